// PointNetSetAbstraction_77661598646511
// MI455X (gfx1250) — compile-verified
//
#include <hip/hip_runtime.h>
#include <hip/hip_bf16.h>
#include <math.h>

typedef __attribute__((ext_vector_type(2))) float v2f;
typedef __attribute__((ext_vector_type(8))) float v8f;

#define NPTS   4096
#define NSAMP  1024
#define KNN    32
#define BATCH  16
#define MCOLS  524288   // B * NSAMP * KNN = 2^19

// Pair-interleaved activation layout:
//   elem(c, m) lives at  base[(c>>1) * 2*MCOLS + 2*m + (c&1)]
// so channel rows (2r, 2r+1) are interleaved per column -> b64 loads/stores.

// ---------------------------------------------------------------------------
// 1) Farthest point sampling: one block per batch, 256 threads, 1024 steps.
// ---------------------------------------------------------------------------
__global__ void fps_kernel(const float* __restrict__ xyz,
                           const int* __restrict__ init_f,
                           float* __restrict__ new_xyz) {
  const int b   = blockIdx.x;
  const int tid = threadIdx.x;               // 256 threads
  const float* px = xyz + (size_t)b * NPTS * 3;

  __shared__ float s_val[256];
  __shared__ int   s_idx[256];
  __shared__ float s_cent[3];

  float dist[16];
#pragma unroll
  for (int j = 0; j < 16; ++j) dist[j] = 1e10f;

  int farthest = init_f[b];

  for (int i = 0; i < NSAMP; ++i) {
    if (tid == 0) {
      float cx = px[farthest * 3 + 0];
      float cy = px[farthest * 3 + 1];
      float cz = px[farthest * 3 + 2];
      s_cent[0] = cx; s_cent[1] = cy; s_cent[2] = cz;
      float* o = new_xyz + ((size_t)b * NSAMP + i) * 3;
      o[0] = cx; o[1] = cy; o[2] = cz;       // emit BEFORE update (scan order)
    }
    __syncthreads();
    const float cx = s_cent[0], cy = s_cent[1], cz = s_cent[2];

    float bv = -1.0f; int bi = 0;
#pragma unroll
    for (int j = 0; j < 16; ++j) {
      const int n = tid + j * 256;           // coalesced
      const float dx = px[n * 3 + 0] - cx;
      const float dy = px[n * 3 + 1] - cy;
      const float dz = px[n * 3 + 2] - cz;
      float d = dx * dx + dy * dy + dz * dz;
      float dm = dist[j];
      dm = (d < dm) ? d : dm;
      dist[j] = dm;
      if (dm > bv) { bv = dm; bi = n; }      // strict > keeps first occurrence
    }
    s_val[tid] = bv; s_idx[tid] = bi;
    __syncthreads();
    for (int off = 128; off; off >>= 1) {    // argmax, tie -> smaller index
      if (tid < off) {
        float ov = s_val[tid + off]; int oi = s_idx[tid + off];
        if (ov > s_val[tid] || (ov == s_val[tid] && oi < s_idx[tid])) {
          s_val[tid] = ov; s_idx[tid] = oi;
        }
      }
      __syncthreads();
    }
    farthest = s_idx[0];
    __syncthreads();
  }
}

// ---------------------------------------------------------------------------
// 2) KNN among sampled points + grouping (one wave32 per query row).
//    Writes centered groups in pair-interleaved GEMM layout (rows 0..2).
// ---------------------------------------------------------------------------
__global__ void knn_group_kernel(const float* __restrict__ xyz,
                                 const float* __restrict__ new_xyz,
                                 float* __restrict__ x0) {
  __shared__ float s_pts[NSAMP * 3];         // 12 KB
  const int b   = blockIdx.x >> 7;
  const int grp = blockIdx.x & 127;
  const float* nx = new_xyz + (size_t)b * NSAMP * 3;
  for (int i = threadIdx.x; i < NSAMP * 3; i += 256) s_pts[i] = nx[i];
  __syncthreads();

  const int wave = threadIdx.x >> 5;
  const int lane = threadIdx.x & 31;
  const int s = grp * 8 + wave;

  const float qx = s_pts[s * 3 + 0];
  const float qy = s_pts[s * 3 + 1];
  const float qz = s_pts[s * 3 + 2];

  float d[32];
#pragma unroll
  for (int t = 0; t < 32; ++t) {
    const int j = t * 32 + lane;
    const float dx = s_pts[j * 3 + 0] - qx;
    const float dy = s_pts[j * 3 + 1] - qy;
    const float dz = s_pts[j * 3 + 2] - qz;
    d[t] = dx * dx + dy * dy + dz * dz;
  }

  int sel = 0;
  for (int r = 0; r < KNN + 1; ++r) {
    float bv = 3.4e38f; int bj = 1 << 30;
#pragma unroll
    for (int t = 0; t < 32; ++t) {           // j increases with t -> stable
      const int j = t * 32 + lane;
      if (d[t] < bv) { bv = d[t]; bj = j; }
    }
    for (int off = 16; off; off >>= 1) {     // wave argmin, tie -> smaller j
      const float ov = __shfl_xor(bv, off, 32);
      const int   oj = __shfl_xor(bj, off, 32);
      if (ov < bv || (ov == bv && oj < bj)) { bv = ov; bj = oj; }
    }
    if (r >= 1 && lane == (r - 1)) sel = bj; // ranks 1..32
    if ((bj & 31) == lane) {
      const int tt = bj >> 5;
#pragma unroll
      for (int t = 0; t < 32; ++t)           // select-based kill, stays in regs
        if (t == tt) d[t] = 3.4e38f;
    }
  }

  // Reference gathers from RAW xyz at indices < 1024 (replicated exactly).
  const float* src = xyz + ((size_t)b * NPTS + sel) * 3;
  const size_t m = ((size_t)(b * NSAMP + s)) * KNN + lane;
  float2 p01; p01.x = src[0] - qx; p01.y = src[1] - qy;
  ((float2*)x0)[m] = p01;                          // channels 0,1 (pair 0)
  x0[2 * (size_t)MCOLS + 2 * m] = src[2] - qz;     // channel 2 (pair 1, even)
}

// ---------------------------------------------------------------------------
// 3) GEMM via V_WMMA_F32_16X16X4_F32:  Y = W * X + bias, pair layout.
//    One wave computes a 16x64 D strip (4 tiles). Compile-time CIN/COUT.
//    APPLY_BN: fuse previous layer's BN+ReLU into the X loads.
// ---------------------------------------------------------------------------
template <int CIN, int COUT, bool APPLY_BN>
__global__ void gemm_wmma_f32(const float* __restrict__ W,
                              const float* __restrict__ bias,
                              const float* __restrict__ X,
                              float* __restrict__ Y,
                              const float* __restrict__ sc_in,
                              const float* __restrict__ sh_in) {
  const int wave = (blockIdx.x * blockDim.x + threadIdx.x) >> 5;
  const int lane = threadIdx.x & 31;
  constexpr int numRowT = COUT / 16;
  const int colG = wave / numRowT;
  const int rowT = wave - colG * numRowT;
  const size_t n0 = (size_t)colG * 64;             // 4 column tiles per wave
  const int co0 = rowT * 16;
  const int lo = lane & 15;
  const int hi = lane >> 4;

  v8f acc0 = {}, acc1 = {}, acc2 = {}, acc3 = {};
  constexpr int KPAD = (CIN + 3) & ~3;

#pragma unroll
  for (int k0 = 0; k0 < KPAD; k0 += 4) {
    const int ka = k0 + 2 * hi;                    // even
    const int kb = ka + 1;

    v2f a;
    if (CIN % 4 == 0) {
      // rows are float2-aligned; single b64 load
      const float2 wa = ((const float2*)(W + (size_t)(co0 + lo) * CIN + k0))[hi];
      a.x = wa.x; a.y = wa.y;
    } else {                                       // CIN==3 tail-guarded path
      a.x = (ka < CIN) ? W[(size_t)(co0 + lo) * CIN + ka] : 0.0f;
      a.y = (kb < CIN) ? W[(size_t)(co0 + lo) * CIN + kb] : 0.0f;
    }

    const float2* xp = (const float2*)(X + (size_t)(ka >> 1) * 2 * MCOLS);
    float2 x0v = xp[n0 + 0  + lo];
    float2 x1v = xp[n0 + 16 + lo];
    float2 x2v = xp[n0 + 32 + lo];
    float2 x3v = xp[n0 + 48 + lo];

    if (APPLY_BN) {
      const float sa = sc_in[ka], ha = sh_in[ka];
      const float sb = sc_in[kb], hb = sh_in[kb];
      x0v.x = fmaxf(x0v.x * sa + ha, 0.0f); x0v.y = fmaxf(x0v.y * sb + hb, 0.0f);
      x1v.x = fmaxf(x1v.x * sa + ha, 0.0f); x1v.y = fmaxf(x1v.y * sb + hb, 0.0f);
      x2v.x = fmaxf(x2v.x * sa + ha, 0.0f); x2v.y = fmaxf(x2v.y * sb + hb, 0.0f);
      x3v.x = fmaxf(x3v.x * sa + ha, 0.0f); x3v.y = fmaxf(x3v.y * sb + hb, 0.0f);
    }
    if (CIN % 4 != 0) {                            // zero the padded K rows
      if (ka >= CIN) { x0v.x = 0; x1v.x = 0; x2v.x = 0; x3v.x = 0; }
      if (kb >= CIN) { x0v.y = 0; x1v.y = 0; x2v.y = 0; x3v.y = 0; }
    }

    v2f b0; b0.x = x0v.x; b0.y = x0v.y;
    v2f b1; b1.x = x1v.x; b1.y = x1v.y;
    v2f b2; b2.x = x2v.x; b2.y = x2v.y;
    v2f b3; b3.x = x3v.x; b3.y = x3v.y;

    acc0 = __builtin_amdgcn_wmma_f32_16x16x4_f32(false, a, false, b0, (short)0, acc0, false, false);
    acc1 = __builtin_amdgcn_wmma_f32_16x16x4_f32(false, a, false, b1, (short)0, acc1, false, false);
    acc2 = __builtin_amdgcn_wmma_f32_16x16x4_f32(false, a, false, b2, (short)0, acc2, false, false);
    acc3 = __builtin_amdgcn_wmma_f32_16x16x4_f32(false, a, false, b3, (short)0, acc3, false, false);
  }

  // Store: D rows (v, v+1) are a channel pair -> one b64 per pair per tile.
#pragma unroll
  for (int v = 0; v < 8; v += 2) {
    const int m0 = co0 + v + 8 * hi;               // even channel of the pair
    const float bx = bias[m0], by = bias[m0 + 1];
    float2* yp = (float2*)(Y + (size_t)(m0 >> 1) * 2 * MCOLS);
    float2 s0; s0.x = acc0[v] + bx; s0.y = acc0[v + 1] + by;
    float2 s1; s1.x = acc1[v] + bx; s1.y = acc1[v + 1] + by;
    float2 s2; s2.x = acc2[v] + bx; s2.y = acc2[v + 1] + by;
    float2 s3; s3.x = acc3[v] + bx; s3.y = acc3[v + 1] + by;
    yp[n0 + 0  + lo] = s0;
    yp[n0 + 16 + lo] = s1;
    yp[n0 + 32 + lo] = s2;
    yp[n0 + 48 + lo] = s3;
  }
}

// ---------------------------------------------------------------------------
// 4) BN statistics on pair layout: block reduces one chunk of one channel
//    PAIR with coalesced b64 reads. Deterministic two-stage reduction.
// ---------------------------------------------------------------------------
__global__ void bnstat_pair(const float* __restrict__ Y,
                            float* __restrict__ psum,
                            float* __restrict__ psumsq) {
  const int cp = blockIdx.y, chunk = blockIdx.x;   // 128 chunks of 4096 cols
  const float2* p = (const float2*)(Y + (size_t)cp * 2 * MCOLS) +
                    (size_t)chunk * 4096;
  float s0 = 0, q0 = 0, s1 = 0, q1 = 0;
  for (int i = threadIdx.x; i < 4096; i += 256) {
    const float2 v = p[i];
    s0 += v.x; q0 += v.x * v.x;
    s1 += v.y; q1 += v.y * v.y;
  }
  __shared__ float l0[256], m0[256], l1[256], m1[256];
  l0[threadIdx.x] = s0; m0[threadIdx.x] = q0;
  l1[threadIdx.x] = s1; m1[threadIdx.x] = q1;
  __syncthreads();
  for (int off = 128; off; off >>= 1) {
    if (threadIdx.x < off) {
      l0[threadIdx.x] += l0[threadIdx.x + off];
      m0[threadIdx.x] += m0[threadIdx.x + off];
      l1[threadIdx.x] += l1[threadIdx.x + off];
      m1[threadIdx.x] += m1[threadIdx.x + off];
    }
    __syncthreads();
  }
  if (threadIdx.x == 0) {
    const int c0 = 2 * cp;
    psum[c0 * 128 + chunk]         = l0[0];
    psumsq[c0 * 128 + chunk]       = m0[0];
    psum[(c0 + 1) * 128 + chunk]   = l1[0];
    psumsq[(c0 + 1) * 128 + chunk] = m1[0];
  }
}

__global__ void bnstat_final(const float* __restrict__ psum,
                             const float* __restrict__ psumsq,
                             const float* __restrict__ g,
                             const float* __restrict__ be,
                             float* __restrict__ scale,
                             float* __restrict__ shift) {
  const int c = blockIdx.x;                  // 128 threads per block
  __shared__ float ls[128], ls2[128];
  ls[threadIdx.x]  = psum[c * 128 + threadIdx.x];
  ls2[threadIdx.x] = psumsq[c * 128 + threadIdx.x];
  __syncthreads();
  for (int off = 64; off; off >>= 1) {
    if (threadIdx.x < off) {
      ls[threadIdx.x]  += ls[threadIdx.x + off];
      ls2[threadIdx.x] += ls2[threadIdx.x + off];
    }
    __syncthreads();
  }
  if (threadIdx.x == 0) {
    const float inv = 1.0f / (float)MCOLS;
    const float m = ls[0] * inv;
    const float var = ls2[0] * inv - m * m;  // biased var (matches jnp.var)
    const float sc = g[c] * rsqrtf(var + 1e-5f);
    scale[c] = sc;
    shift[c] = be[c] - m * sc;
  }
}

// ---------------------------------------------------------------------------
// 5) Fused BN3 + ReLU + max over K. Each thread handles one channel PAIR at
//    one (b, s): 32 contiguous float2 reads, two outputs.
// ---------------------------------------------------------------------------
__global__ void maxk_kernel(const float* __restrict__ Y3,
                            const float* __restrict__ scale,
                            const float* __restrict__ shift,
                            float* __restrict__ out) {
  const int t = blockIdx.x * 256 + threadIdx.x;   // (b*64 + cp)*1024 + s
  const int s  = t & 1023;
  const int cp = (t >> 10) & 63;
  const int b  = t >> 16;
  const int c0 = 2 * cp;
  const size_t mbase = (size_t)((b << 10) | s) << 5;
  const float2* p = (const float2*)(Y3 + (size_t)cp * 2 * MCOLS) + mbase;
  const float sa = scale[c0], ha = shift[c0];
  const float sb = scale[c0 + 1], hb = shift[c0 + 1];
  float mx = 0.0f, my = 0.0f;                     // relu outputs are >= 0
#pragma unroll
  for (int k = 0; k < 32; ++k) {
    const float2 v = p[k];
    mx = fmaxf(mx, fmaxf(v.x * sa + ha, 0.0f));
    my = fmaxf(my, fmaxf(v.y * sb + hb, 0.0f));
  }
  out[(size_t)((b * 128 + c0) << 10) + s]     = mx;
  out[(size_t)((b * 128 + c0 + 1) << 10) + s] = my;
}

// ---------------------------------------------------------------------------
extern "C" void kernel_launch(void* const* d_in, const int* in_sizes, int n_in,
                              void* d_out, int out_size, void* d_ws, size_t ws_size,
                              hipStream_t stream) {
  (void)in_sizes; (void)n_in; (void)out_size; (void)ws_size;
  const float* xyz    = (const float*)d_in[0];
  const int*   init_f = (const int*)d_in[1];
  const float* W1 = (const float*)d_in[2];
  const float* b1 = (const float*)d_in[3];
  const float* g1 = (const float*)d_in[4];
  const float* be1 = (const float*)d_in[5];
  const float* W2 = (const float*)d_in[6];
  const float* b2 = (const float*)d_in[7];
  const float* g2 = (const float*)d_in[8];
  const float* be2 = (const float*)d_in[9];
  const float* W3 = (const float*)d_in[10];
  const float* b3 = (const float*)d_in[11];
  const float* g3 = (const float*)d_in[12];
  const float* be3 = (const float*)d_in[13];

  float* out      = (float*)d_out;
  float* new_xyz  = out;                       // 16*1024*3 = 49152 floats
  float* new_pts  = out + 49152;               // 16*128*1024 floats

  const size_t M = MCOLS;
  float* ws = (float*)d_ws;
  float* R3 = ws;                              // 128*M (y3; y1 in low half)
  float* R2 = ws + 128 * M;                    // 64*M  (y2)
  float* X0 = ws + 192 * M;                    // 4*M   (rows 0..2 + pad row 3)
  float* PS  = ws + 196 * M;                   // 128ch * 128chunks
  float* PS2 = PS + 128 * 128;
  float* SC  = PS2 + 128 * 128;                // 128
  float* SH  = SC + 128;                       // 128

  // 1) FPS -> new_xyz
  fps_kernel<<<BATCH, 256, 0, stream>>>(xyz, init_f, new_xyz);

  // 2) KNN + grouping -> X0 (3 x M, pair layout)
  knn_group_kernel<<<BATCH * 128, 256, 0, stream>>>(xyz, new_xyz, X0);

  // Wave/grid math: waves = (M/64) * (COUT/16); 8 waves per block.
  const int blocks64  = (int)((M / 64) * 4 / 8);   // COUT=64  -> 4096 blocks
  const int blocks128 = (int)((M / 64) * 8 / 8);   // COUT=128 -> 8192 blocks

  // 3) Layer 1: (64x3)*(3xM) -> R3 low half; stats
  gemm_wmma_f32<3, 64, false><<<blocks64, 256, 0, stream>>>(W1, b1, X0, R3, nullptr, nullptr);
  bnstat_pair<<<dim3(128, 32), 256, 0, stream>>>(R3, PS, PS2);
  bnstat_final<<<64, 128, 0, stream>>>(PS, PS2, g1, be1, SC, SH);

  // 4) Layer 2: BN1+ReLU fused into loads; -> R2; stats
  gemm_wmma_f32<64, 64, true><<<blocks64, 256, 0, stream>>>(W2, b2, R3, R2, SC, SH);
  bnstat_pair<<<dim3(128, 32), 256, 0, stream>>>(R2, PS, PS2);
  bnstat_final<<<64, 128, 0, stream>>>(PS, PS2, g2, be2, SC, SH);

  // 5) Layer 3: BN2+ReLU fused into loads; -> R3 full (overwrites dead y1)
  gemm_wmma_f32<64, 128, true><<<blocks128, 256, 0, stream>>>(W3, b3, R2, R3, SC, SH);
  bnstat_pair<<<dim3(128, 64), 256, 0, stream>>>(R3, PS, PS2);
  bnstat_final<<<128, 128, 0, stream>>>(PS, PS2, g3, be3, SC, SH);

  // 6) Fused BN3 + ReLU + max over K -> new_points
  maxk_kernel<<<(BATCH * 64 * NSAMP) / 256, 256, 0, stream>>>(R3, SC, SH, new_pts);
}